// Conv_layer_39462159515864
// MI455X (gfx1250) — compile-verified
//
#include <hip/hip_runtime.h>
#include <hip/hip_bf16.h>

typedef __attribute__((ext_vector_type(2))) float v2f;
typedef __attribute__((ext_vector_type(8))) float v8f;

#define IN_DIM  256
#define OUT_DIM 256

// ---------------------------------------------------------------------------
// Zero-fill d_out (harness poisons it; atomics in spmm need a zeroed base).
// Also zeroes the trailing recon scalar (int32 0 bit-equals float 0).
// ---------------------------------------------------------------------------
__global__ void zero_kernel(float* __restrict__ out, size_t n) {
    size_t i = (size_t)blockIdx.x * blockDim.x + threadIdx.x;
    size_t stride = (size_t)gridDim.x * blockDim.x;
    for (; i < n; i += stride) out[i] = 0.0f;
}

// ---------------------------------------------------------------------------
// GEMM: h[100000,256] = x[100000,256] @ W[256,256] + bias, fp32 via
// V_WMMA_F32_16X16X4_F32.  Block = 256 threads = 8 waves.
// Wave w: row-tile = blockIdx*2 + w/4 (16 rows), col strip = (w%4)*64.
// Each wave keeps 4 accumulators (16x16 each) and loops K in steps of 4.
// ---------------------------------------------------------------------------
__device__ inline void store_tile(float* __restrict__ h,
                                  const float* __restrict__ bias,
                                  v8f acc, int row0, int mhi, int col) {
    float bv = bias[col];
#pragma unroll
    for (int r = 0; r < 8; ++r)
        h[(size_t)(row0 + mhi + r) * OUT_DIM + col] = acc[r] + bv;
}

__global__ __launch_bounds__(256) void gemm_kernel(
        const float* __restrict__ x,
        const float* __restrict__ w,
        const float* __restrict__ bias,
        float* __restrict__ h) {
    const int lane = threadIdx.x & 31;
    const int wave = threadIdx.x >> 5;

    const int rowTile = blockIdx.x * 2 + (wave >> 2);
    const int row0    = rowTile * 16;
    const int n0      = (wave & 3) * 64;

    const int m  = lane & 15;        // A: row M / B: col N / C: col N
    const int kg = (lane >> 4) * 2;  // upper half-wave handles K+2,K+3

    v8f acc0 = {}, acc1 = {}, acc2 = {}, acc3 = {};

    const float* xrow = x + (size_t)(row0 + m) * IN_DIM + kg;

#pragma unroll 4
    for (int k = 0; k < IN_DIM; k += 4) {
        // A fragment: x[row0+m][k+kg .. k+kg+1]  (contiguous float2)
        v2f a = *(const v2f*)(xrow + k);

        // B fragments: W[k+kg][n] , W[k+kg+1][n]   n = n0 + nt*16 + m
        const float* wk = w + (size_t)(k + kg) * OUT_DIM;
        v2f b0 = { wk[n0 +  0 + m], wk[OUT_DIM + n0 +  0 + m] };
        v2f b1 = { wk[n0 + 16 + m], wk[OUT_DIM + n0 + 16 + m] };
        v2f b2 = { wk[n0 + 32 + m], wk[OUT_DIM + n0 + 32 + m] };
        v2f b3 = { wk[n0 + 48 + m], wk[OUT_DIM + n0 + 48 + m] };

        acc0 = __builtin_amdgcn_wmma_f32_16x16x4_f32(false, a, false, b0,
                                                     (short)0, acc0, false, false);
        acc1 = __builtin_amdgcn_wmma_f32_16x16x4_f32(false, a, false, b1,
                                                     (short)0, acc1, false, false);
        acc2 = __builtin_amdgcn_wmma_f32_16x16x4_f32(false, a, false, b2,
                                                     (short)0, acc2, false, false);
        acc3 = __builtin_amdgcn_wmma_f32_16x16x4_f32(false, a, false, b3,
                                                     (short)0, acc3, false, false);
    }

    // C/D layout: vgpr r, lane l -> M = r + 8*(l>=16), N = l&15
    const int mhi = (lane >> 4) * 8;
    store_tile(h, bias, acc0, row0, mhi, n0 +  0 + m);
    store_tile(h, bias, acc1, row0, mhi, n0 + 16 + m);
    store_tile(h, bias, acc2, row0, mhi, n0 + 32 + m);
    store_tile(h, bias, acc3, row0, mhi, n0 + 48 + m);
}

// ---------------------------------------------------------------------------
// SpMM scatter: out[rows[e]] += vals[e] * h[cols[e]]   (256 features/edge)
// One wave per edge: 2x float4 per lane covers 256 floats coalesced,
// then 8 non-returning global f32 atomics per lane.  h gather stays in L2
// (100 MB table vs 192 MB L2).
// ---------------------------------------------------------------------------
__global__ __launch_bounds__(256) void spmm_kernel(
        const int* __restrict__ rows,
        const int* __restrict__ cols,
        const float* __restrict__ vals,
        const float* __restrict__ h,
        float* __restrict__ out,
        int nEdges) {
    const int wid  = blockIdx.x * 8 + (threadIdx.x >> 5);
    const int lane = threadIdx.x & 31;
    if (wid >= nEdges) return;

    const int   r = rows[wid];
    const int   c = cols[wid];
    const float v = vals[wid];

    const float4* hp = (const float4*)(h + (size_t)c * OUT_DIM);
    float4 a = hp[lane];        // features [lane*4     .. lane*4+3]
    float4 b = hp[lane + 32];   // features [128+lane*4 .. 128+lane*4+3]

    float* op = out + (size_t)r * OUT_DIM + lane * 4;
    atomicAdd(op + 0, v * a.x);
    atomicAdd(op + 1, v * a.y);
    atomicAdd(op + 2, v * a.z);
    atomicAdd(op + 3, v * a.w);
    float* op2 = op + 128;
    atomicAdd(op2 + 0, v * b.x);
    atomicAdd(op2 + 1, v * b.y);
    atomicAdd(op2 + 2, v * b.z);
    atomicAdd(op2 + 3, v * b.w);
}

// ---------------------------------------------------------------------------
extern "C" void kernel_launch(void* const* d_in, const int* in_sizes, int n_in,
                              void* d_out, int out_size, void* d_ws, size_t ws_size,
                              hipStream_t stream) {
    const float* x      = (const float*)d_in[0];
    const int*   G_rows = (const int*)  d_in[1];
    const int*   G_cols = (const int*)  d_in[2];
    const float* G_vals = (const float*)d_in[3];
    const float* weight = (const float*)d_in[4];
    const float* bias   = (const float*)d_in[5];

    float* out = (float*)d_out;
    float* h   = (float*)d_ws;   // 100000*256*4 = 102.4 MB staging for x@W+b

    const int nNodes = in_sizes[0] / IN_DIM;  // 100000
    const int nEdges = in_sizes[1];           // 3200000

    // 1) zero the output (and recon scalar)
    zero_kernel<<<4096, 256, 0, stream>>>(out, (size_t)out_size);

    // 2) dense transform into workspace: 32 rows per block (2 row-tiles x 8 waves)
    gemm_kernel<<<nNodes / 32, 256, 0, stream>>>(x, weight, bias, h);

    // 3) edge scatter: one wave per edge
    spmm_kernel<<<(nEdges + 7) / 8, 256, 0, stream>>>(G_rows, G_cols, G_vals,
                                                      h, out, nEdges);
}